// GNNBasicBlock_63084479644214
// MI455X (gfx1250) — compile-verified
//
#include <hip/hip_runtime.h>
#include <hip/hip_bf16.h>

// ---------------------------------------------------------------------------
// GNN basic block for MI455X (gfx1250, wave32):
//   conv1 (GEMM via V_WMMA_F32_16X16X4_F32 + edge scatter) -> BN -> lrelu ->
//   conv2 (same) -> BN -> +residual -> lrelu
// Edge scatter is the bandwidth bottleneck (~3.3 GB @ 23.3 TB/s); GEMMs use
// the fp32 WMMA path (exact precision, no downcast needed).
// ---------------------------------------------------------------------------

typedef float v2f __attribute__((ext_vector_type(2)));
typedef float v8f __attribute__((ext_vector_type(8)));

#define D 128
#define NEG_SLOPE 0.01f
#define BN_EPS 1e-5f

static inline unsigned cdiv(long a, long b) { return (unsigned)((a + b - 1) / b); }

// ---------------- utility: zero a float region ------------------------------
__global__ void k_zero(float* __restrict__ p, int n) {
  int i = blockIdx.x * blockDim.x + threadIdx.x;
  for (; i < n; i += gridDim.x * blockDim.x) p[i] = 0.0f;
}

// ---------------- degree count: deg[dst] += 1 -------------------------------
__global__ void k_deg(const int* __restrict__ dstI, float* __restrict__ deg, int E) {
  int i = blockIdx.x * blockDim.x + threadIdx.x;
  if (i < E) atomicAdd(&deg[dstI[i]], 1.0f);
}

// ---------------- dinv = rsqrt(deg + 1)  (self loop) ------------------------
__global__ void k_rsqrt(float* __restrict__ deg, int N) {
  int i = blockIdx.x * blockDim.x + threadIdx.x;
  if (i < N) deg[i] = rsqrtf(deg[i] + 1.0f);
}

// ---------------- fp32 WMMA GEMM: C[M,128] = A[M,128] @ B[128,128] ----------
// One wave computes one 16x16 tile of C; 8 waves/block cover the 128 columns
// of a 16-row stripe.  K-loop: 32 x v_wmma_f32_16x16x4_f32.
// A-operand (16x4, MxK): lane<16 holds row=lane, {K=k,k+1}; lane>=16 holds
// row=lane-16, {K=k+2,k+3}.  B-operand (4x16, KxN) symmetric with N=lane&15.
// C/D: VGPR v -> M = v + (lane>=16 ? 8 : 0), N = lane&15.
__global__ __launch_bounds__(256)
void k_gemm_wmma(const float* __restrict__ A, const float* __restrict__ B,
                 float* __restrict__ C, int M) {
  const int lane = threadIdx.x & 31;
  const int wv   = threadIdx.x >> 5;        // 0..7 -> column tile
  const int r    = lane & 15;
  const int kb   = (lane >> 4) << 1;        // 0 or 2
  const int row0 = blockIdx.x * 16;
  const int col0 = wv * 16;

  int ar = row0 + r;                        // clamp reads for tail tile; WMMA
  if (ar > M - 1) ar = M - 1;               // needs EXEC all-ones
  const float* ap = A + (size_t)ar * D + kb;
  const float* bp = B + (size_t)kb * D + col0 + r;

  v8f acc = {};
  for (int k = 0; k < D; k += 4) {
    v2f a, b;
    a.x = ap[k];
    a.y = ap[k + 1];
    b.x = bp[(size_t)k * D];
    b.y = bp[(size_t)(k + 1) * D];
    acc = __builtin_amdgcn_wmma_f32_16x16x4_f32(
        /*neg_a=*/false, a, /*neg_b=*/false, b,
        /*c_mod=*/(short)0, acc, /*reuse_a=*/false, /*reuse_b=*/false);
  }

  const int mrow = row0 + ((lane >> 4) << 3);
  float* cp = C + (size_t)mrow * D + col0 + (lane & 15);
#pragma unroll
  for (int v = 0; v < 8; ++v)
    if (mrow + v < M) cp[(size_t)v * D] = acc[v];
}

// ---------------- agg init with self-loop term: agg = h * dinv^2 ------------
__global__ void k_selfscale(const float* __restrict__ h, const float* __restrict__ dinv,
                            float* __restrict__ agg, int N) {
  int t = blockIdx.x * blockDim.x + threadIdx.x;      // N*32 float4 slots
  if (t >= N * (D / 4)) return;
  int row = t >> 5;
  float di = dinv[row];
  float s = di * di;
  const float4 v = reinterpret_cast<const float4*>(h)[t];
  float4 o;
  o.x = v.x * s; o.y = v.y * s; o.z = v.z * s; o.w = v.w * s;
  reinterpret_cast<float4*>(agg)[t] = o;
}

// ---------------- edge scatter: agg[dst] += h[src] * dinv[src]*dinv[dst] ----
// One wave per edge: 32 lanes x float4 gather + 4 global f32 atomics each.
__global__ __launch_bounds__(256)
void k_scatter(const int* __restrict__ srcI, const int* __restrict__ dstI,
               const float* __restrict__ h, const float* __restrict__ dinv,
               float* __restrict__ agg, int E) {
  int gid  = blockIdx.x * blockDim.x + threadIdx.x;
  int e    = gid >> 5;
  int lane = gid & 31;
  if (e >= E) return;
  int s = srcI[e];
  int d = dstI[e];
  float w = dinv[s] * dinv[d];
  const float4 v = *reinterpret_cast<const float4*>(h + (size_t)s * D + lane * 4);
  float* out = agg + (size_t)d * D + lane * 4;
  atomicAdd(out + 0, v.x * w);
  atomicAdd(out + 1, v.y * w);
  atomicAdd(out + 2, v.z * w);
  atomicAdd(out + 3, v.w * w);
}

// ---------------- BN stats: per-column sum / sumsq --------------------------
__global__ __launch_bounds__(128)
void k_bnstats(const float* __restrict__ X, float* __restrict__ sums, int N) {
  const int ROWS = 128;
  int c  = threadIdx.x;                    // column 0..127
  long r0 = (long)blockIdx.x * ROWS;
  long r1 = r0 + ROWS; if (r1 > N) r1 = N;
  float s = 0.0f, s2 = 0.0f;
  for (long r = r0; r < r1; ++r) {
    float v = X[r * D + c];
    s += v; s2 += v * v;
  }
  atomicAdd(&sums[c], s);
  atomicAdd(&sums[D + c], s2);
}

// ---------------- BN coefficients: scale/shift per column -------------------
// (conv bias cancels exactly under BN mean subtraction, so it is omitted)
__global__ void k_bncoef(const float* __restrict__ sums, const float* __restrict__ g,
                         const float* __restrict__ be, float* __restrict__ coef,
                         float invN) {
  int c = threadIdx.x;
  float mean = sums[c] * invN;
  float var  = sums[D + c] * invN - mean * mean;
  float scale = g[c] * rsqrtf(var + BN_EPS);
  coef[c]     = scale;
  coef[D + c] = be[c] - mean * scale;
}

// ---------------- y = lrelu(BN(x)) ------------------------------------------
__global__ void k_bnapply_lrelu(const float* __restrict__ X, const float* __restrict__ coef,
                                float* __restrict__ Y, int total) {
  int i = blockIdx.x * blockDim.x + threadIdx.x;
  if (i >= total) return;
  int c = i & (D - 1);
  float v = X[i] * coef[c] + coef[D + c];
  Y[i] = v > 0.0f ? v : NEG_SLOPE * v;
}

// ---------------- out = lrelu(BN(out) + x)  (in place) ----------------------
__global__ void k_final(float* __restrict__ Y, const float* __restrict__ coef,
                        const float* __restrict__ xin, int total) {
  int i = blockIdx.x * blockDim.x + threadIdx.x;
  if (i >= total) return;
  int c = i & (D - 1);
  float v = Y[i] * coef[c] + coef[D + c] + xin[i];
  Y[i] = v > 0.0f ? v : NEG_SLOPE * v;
}

// ---------------------------------------------------------------------------
extern "C" void kernel_launch(void* const* d_in, const int* in_sizes, int n_in,
                              void* d_out, int out_size, void* d_ws, size_t ws_size,
                              hipStream_t stream) {
  const float* x   = (const float*)d_in[0];
  const int*   ei  = (const int*)d_in[1];
  const float* W1  = (const float*)d_in[2];
  // d_in[3] = b1 : cancels under BatchNorm (added then mean-subtracted)
  const float* g1  = (const float*)d_in[4];
  const float* be1 = (const float*)d_in[5];
  const float* W2  = (const float*)d_in[6];
  // d_in[7] = b2 : cancels under BatchNorm
  const float* g2  = (const float*)d_in[8];
  const float* be2 = (const float*)d_in[9];

  const int N  = in_sizes[0] / D;
  const int E  = in_sizes[1] / 2;
  const long ND = (long)N * D;
  const int* srcI = ei;
  const int* dstI = ei + E;

  float* out = (float*)d_out;          // also used as big scratch (h/agg ping)
  float* agg   = (float*)d_ws;         // N*D scratch
  float* dinv  = agg + ND;             // N
  float* sums1 = dinv + N;             // 2*D
  float* coef1 = sums1 + 2 * D;        // 2*D
  float* sums2 = coef1 + 2 * D;        // 2*D
  float* coef2 = sums2 + 2 * D;        // 2*D

  const int nzero = N + 8 * D;         // dinv + all stats regions
  k_zero<<<cdiv(nzero, 256), 256, 0, stream>>>(dinv, nzero);

  // degrees -> dinv = rsqrt(deg+1)
  k_deg<<<cdiv(E, 256), 256, 0, stream>>>(dstI, dinv, E);
  k_rsqrt<<<cdiv(N, 256), 256, 0, stream>>>(dinv, N);

  // ---- layer 1 ----
  k_gemm_wmma<<<cdiv(N, 16), 256, 0, stream>>>(x, W1, out, N);        // h1 -> out
  k_selfscale<<<cdiv((long)N * 32, 256), 256, 0, stream>>>(out, dinv, agg, N);
  k_scatter<<<cdiv((long)E * 32, 256), 256, 0, stream>>>(srcI, dstI, out, dinv, agg, E);
  k_bnstats<<<cdiv(N, 128), 128, 0, stream>>>(agg, sums1, N);
  k_bncoef<<<1, D, 0, stream>>>(sums1, g1, be1, coef1, 1.0f / (float)N);
  k_bnapply_lrelu<<<cdiv(ND, 256), 256, 0, stream>>>(agg, coef1, out, (int)ND); // y1 -> out

  // ---- layer 2 ----
  k_gemm_wmma<<<cdiv(N, 16), 256, 0, stream>>>(out, W2, agg, N);      // h2 -> agg
  k_selfscale<<<cdiv((long)N * 32, 256), 256, 0, stream>>>(agg, dinv, out, N);
  k_scatter<<<cdiv((long)E * 32, 256), 256, 0, stream>>>(srcI, dstI, agg, dinv, out, E);
  k_bnstats<<<cdiv(N, 128), 128, 0, stream>>>(out, sums2, N);
  k_bncoef<<<1, D, 0, stream>>>(sums2, g2, be2, coef2, 1.0f / (float)N);
  k_final<<<cdiv(ND, 256), 256, 0, stream>>>(out, coef2, x, (int)ND);
}